// PagedAttention_35201551958133
// MI455X (gfx1250) — compile-verified
//
#include <hip/hip_runtime.h>
#include <math.h>

typedef __attribute__((ext_vector_type(2))) float v2f;
typedef __attribute__((ext_vector_type(8))) float v8f;
typedef __attribute__((ext_vector_type(4))) unsigned int u32x4;
typedef __attribute__((ext_vector_type(8))) int i32x8;
typedef __attribute__((ext_vector_type(4))) int i32x4;

constexpr int kB   = 2;
constexpr int kS   = 256;
constexpr int kE   = 512;
constexpr int kH   = 8;
constexpr int kDH  = 64;
constexpr int kTC  = 768;
constexpr int kTCH = kTC * kH;          // 6144 cached rows per batch
constexpr int kL   = kTCH + kS * kH;    // 8192 total K/V rows per batch
constexpr int kPad = 68;                // attn LDS row stride (TDM pads 64->68)
constexpr int kGP  = 20;                // gemm LDS row stride (TDM pads 16->20)

// fp32 tensor op: D(16x16) = A(16x4) * B(4x16) + C
__device__ __forceinline__ v8f wmma4(v2f a, v2f b, v8f c) {
  return __builtin_amdgcn_wmma_f32_16x16x4_f32(false, a, false, b, (short)0, c,
                                               false, false);
}

// XOR-swizzle within 32 lanes (and=0x1f, or=0, xor=m); masks 1/2/4/8 stay
// inside each 16-lane half, matching the C/D fragment row grouping.
#define SWZ(x, m)                                                              \
  __int_as_float(__builtin_amdgcn_ds_swizzle(__float_as_int(x),                \
                                             (((m) << 10) | 0x1f)))

// ---------------------------------------------------------------------------
// TDM: async DMA of a rows x elems fp32 tile (row stride stride_elems) from
// global into LDS, padded +4 DWORDs every (1<<pi_code)*2 DWORDs (pi_code must
// match elems so one pad lands per row).  Tracked by TENSORcnt.
// D# packing per CDNA5 ISA 8.3/8.4; workgroup_mask=0 (not in a cluster).
// 6-arg builtin on this toolchain; groups 2/3 + extra group zero (2-D tile).
// ---------------------------------------------------------------------------
__device__ __forceinline__ void tdm_load_2d(const float* gsrc,
                                            unsigned int lds_off, int rows,
                                            int elems, int stride_elems,
                                            int pi_code) {
  const unsigned long long ga = (unsigned long long)(uintptr_t)gsrc;
  u32x4 g0;
  g0[0] = 1u;                                    // count=1, user descriptor
  g0[1] = lds_off;                               // LDS byte address
  g0[2] = (unsigned int)ga;                      // global_addr[31:0]
  g0[3] = (unsigned int)(ga >> 32) | (2u << 30); // global_addr[56:32] | type=2
  i32x8 g1;
  g1[0] = (2 << 16) | (1 << 20) | (pi_code << 22) | (3 << 25);
  //       data=4B    pad_enable   pad_interval      pad_amount=4dw
  g1[1] = elems << 16;    // tensor_dim0 (elements)
  g1[2] = rows << 16;     // tensor_dim1 (== tile rows, no OOB)
  g1[3] = elems << 16;    // tile_dim0
  g1[4] = rows;           // tile_dim1 (tile_dim2 = 0)
  g1[5] = stride_elems;   // tensor_dim0_stride (low 32 of 48)
  g1[6] = 0;
  g1[7] = 0;
  const i32x4 z4 = {0, 0, 0, 0};
  const i32x8 z8 = {0, 0, 0, 0, 0, 0, 0, 0};
  __builtin_amdgcn_tensor_load_to_lds(g0, g1, z4, z4, z8, 0);
}

// ---------------------------------------------------------------------------
// C[M][N] = A[M][K] @ W[N][K]^T + bias[N]   (torch Linear)
// TDM-fed: per 16-K chunk, two async DMAs (A tile 16x16, W tile 64x16, both
// strided by K) into double-buffered LDS; one s_wait_tensorcnt per chunk;
// fragments read from LDS at DS latency (stride-20 rows => conflict-free).
// One wave computes a 16(M) x 64(N) strip with 4 independent WMMA chains.
// ---------------------------------------------------------------------------
__global__ void __launch_bounds__(128)
gemm_xwt_wmma(const float* __restrict__ A, const float* __restrict__ W,
              const float* __restrict__ bias, float* __restrict__ C,
              int M, int N, int K) {
  __shared__ float atile[4][2][16 * kGP];   // [wave][buf]
  __shared__ float wtile[4][2][64 * kGP];

  const int lane  = threadIdx.x & 31;
  const int waveu = __builtin_amdgcn_readfirstlane((int)(threadIdx.x >> 5));
  const int hi    = lane >> 4;
  const int hi2   = hi << 1;
  const int l16   = lane & 15;
  const int m0    = (blockIdx.x * 4 + waveu) * 16;
  const int n0    = blockIdx.y * 64;
  if (m0 >= M || n0 >= N) return;   // wave-uniform guard

  unsigned int aoff[2], woff[2];
  aoff[0] = (unsigned int)(uintptr_t)&atile[waveu][0][0];
  aoff[1] = (unsigned int)(uintptr_t)&atile[waveu][1][0];
  woff[0] = (unsigned int)(uintptr_t)&wtile[waveu][0][0];
  woff[1] = (unsigned int)(uintptr_t)&wtile[waveu][1][0];

  const float* Abase = A + (size_t)m0 * K;
  const float* Wbase = W + (size_t)n0 * K;

  // Prime the pipeline: chunk 0 into buffer 0.
  tdm_load_2d(Abase, aoff[0], 16, 16, K, 3);
  tdm_load_2d(Wbase, woff[0], 64, 16, K, 3);

  v8f acc[4] = {};
  int n = 0;
#pragma unroll 1
  for (int k0 = 0; k0 < K; k0 += 16) {
    const int kn = (k0 + 16 < K) ? (k0 + 16) : 0;   // dummy reload at the end
    tdm_load_2d(Abase + kn, aoff[n ^ 1], 16, 16, K, 3);
    tdm_load_2d(Wbase + kn, woff[n ^ 1], 64, 16, K, 3);
    __builtin_amdgcn_s_wait_tensorcnt(2);   // older pair done; new pair flies

    const float* at = &atile[waveu][n][0];
    const float* wt = &wtile[waveu][n][0];
    v2f a[4], bf[4][4];
#pragma unroll
    for (int ks = 0; ks < 4; ++ks) {
      const int kk = ks * 4 + hi2;
      const float* ar = at + l16 * kGP + kk;
      a[ks].x = ar[0];
      a[ks].y = ar[1];
#pragma unroll
      for (int t = 0; t < 4; ++t) {
        const float* wr = wt + (t * 16 + l16) * kGP + kk;
        bf[ks][t].x = wr[0];
        bf[ks][t].y = wr[1];
      }
    }
#pragma unroll
    for (int ks = 0; ks < 4; ++ks) {
#pragma unroll
      for (int t = 0; t < 4; ++t) acc[t] = wmma4(a[ks], bf[ks][t], acc[t]);
    }
    n ^= 1;
  }

  const float bv0 = bias[n0 + l16];
  const float bv1 = bias[n0 + 16 + l16];
  const float bv2 = bias[n0 + 32 + l16];
  const float bv3 = bias[n0 + 48 + l16];
#pragma unroll
  for (int v = 0; v < 8; ++v) {
    const int r = m0 + v + hi * 8;
    float* Crow = C + (size_t)r * N + n0 + l16;
    Crow[0]  = acc[0][v] + bv0;
    Crow[16] = acc[1][v] + bv1;
    Crow[32] = acc[2][v] + bv2;
    Crow[48] = acc[3][v] + bv3;
  }
}

// ---------------------------------------------------------------------------
// Build contiguous K/V [B, L, DH] = cache rows then new rows (token-major).
// ---------------------------------------------------------------------------
__global__ void __launch_bounds__(256)
repack_kv(const float* __restrict__ kc, const float* __restrict__ vc,
          const float* __restrict__ yk, const float* __restrict__ yv,
          float* __restrict__ Kf, float* __restrict__ Vf) {
  const int idx4 = blockIdx.x * blockDim.x + threadIdx.x;
  const int total4 = kB * kL * (kDH / 4);
  if (idx4 >= total4) return;
  const int d4 = (idx4 % (kDH / 4)) * 4;
  const int l  = (idx4 / (kDH / 4)) % kL;
  const int b  = idx4 / ((kDH / 4) * kL);
  size_t src;
  const float4 *ks, *vs;
  if (l < kTCH) {
    src = ((size_t)b * kTCH + l) * kDH + d4;
    ks = (const float4*)(kc + src);
    vs = (const float4*)(vc + src);
  } else {
    const int t = l - kTCH;
    const int s = t / kH;
    const int h = t % kH;
    src = ((size_t)(b * kS + s)) * kE + h * kDH + d4;
    ks = (const float4*)(yk + src);
    vs = (const float4*)(yv + src);
  }
  ((float4*)Kf)[idx4] = *ks;
  ((float4*)Vf)[idx4] = *vs;
}

// ---------------------------------------------------------------------------
// Flash attention, TDM-fed: one wave per (b, h, 16-query tile).  K/V tiles
// stream through double-buffered LDS via tensor_load_to_lds; the wave pays one
// s_wait_tensorcnt per chunk and builds WMMA fragments from LDS (stride-68
// rows => conflict-free).  2 waves/block keeps static LDS at 37 KB.
// ---------------------------------------------------------------------------
#define LW 17  // P-tile LDS row stride

__device__ __forceinline__ void attn_chunk_lds(const float* kt, const float* vt,
                                               const v2f qa[16],
                                               float* __restrict__ pw, int hi,
                                               int l16, v8f acc[4], float mi[8],
                                               float li[8]) {
  const int hi2 = hi << 1;

  // K B-frags from LDS: B[k][n] = K[l0+n][k]; lane n = l16, k = ks*4+hi2
  v2f kb[16];
#pragma unroll
  for (int ks = 0; ks < 16; ++ks) {
    const float* kr = kt + l16 * kPad + ks * 4 + hi2;
    kb[ks].x = kr[0];
    kb[ks].y = kr[1];
  }

  // ---- S = Q*K^T as two independent WMMA chains
  v8f sA = {}, sB = {};
#pragma unroll
  for (int ks = 0; ks < 16; ks += 2) {
    sA = wmma4(qa[ks], kb[ks], sA);
    sB = wmma4(qa[ks + 1], kb[ks + 1], sB);
  }

  // V B-frags from LDS: B[k][n] = V[l0+k][n]; row = kk, col = t*16+l16
  v2f vb[4][4];
#pragma unroll
  for (int ks2 = 0; ks2 < 4; ++ks2) {
    const int kk = ks2 * 4 + hi2;
    const float* vr = vt + kk * kPad + l16;
#pragma unroll
    for (int t = 0; t < 4; ++t) {
      vb[ks2][t].x = vr[t * 16];
      vb[ks2][t].y = vr[kPad + t * 16];
    }
  }

  // ---- online softmax per row (16 lanes of a half hold one row)
#pragma unroll
  for (int v = 0; v < 8; ++v) {
    const float sv = (sA[v] + sB[v]) * 0.125f;   // 1/sqrt(DH)
    float mc = sv;
    mc = fmaxf(mc, SWZ(mc, 1));
    mc = fmaxf(mc, SWZ(mc, 2));
    mc = fmaxf(mc, SWZ(mc, 4));
    mc = fmaxf(mc, SWZ(mc, 8));
    const float mn   = fmaxf(mi[v], mc);
    const float p    = expf(sv - mn);
    const float corr = expf(mi[v] - mn);         // 0 on first chunk (mi=-inf)
    float rs = p;
    rs += SWZ(rs, 1);
    rs += SWZ(rs, 2);
    rs += SWZ(rs, 4);
    rs += SWZ(rs, 8);
    li[v] = li[v] * corr + rs;
    mi[v] = mn;
    acc[0][v] *= corr;
    acc[1][v] *= corr;
    acc[2][v] *= corr;
    acc[3][v] *= corr;
    pw[(v + hi * 8) * LW + l16] = p;             // P tile, row-major in LDS
  }

  // ---- O += P * V ; P A-frags from LDS (in-order DS, no barrier needed)
#pragma unroll
  for (int ks2 = 0; ks2 < 4; ++ks2) {
    const int kk = ks2 * 4 + hi2;
    v2f pa;
    pa.x = pw[l16 * LW + kk];
    pa.y = pw[l16 * LW + kk + 1];
#pragma unroll
    for (int t = 0; t < 4; ++t) acc[t] = wmma4(pa, vb[ks2][t], acc[t]);
  }
}

__global__ void __launch_bounds__(64)
attn_flash_wmma(const float* __restrict__ yq, const float* __restrict__ Kf,
                const float* __restrict__ Vf, float* __restrict__ O) {
  __shared__ float ktile[2][2][16 * kPad];   // [wave][buf]
  __shared__ float vtile[2][2][16 * kPad];
  __shared__ float plds[2][16 * LW];

  const int lane  = threadIdx.x & 31;
  // Descriptors must be scalar: pin wave id to an SGPR.
  const int waveu = __builtin_amdgcn_readfirstlane((int)(threadIdx.x >> 5));
  const int hi    = lane >> 4;
  const int hi2   = hi << 1;
  const int l16   = lane & 15;
  const int m0    = (blockIdx.x * 2 + waveu) * 16;  // query tile base
  const int h     = blockIdx.y;
  const int b     = blockIdx.z;

  unsigned int koff[2], voff[2];
  koff[0] = (unsigned int)(uintptr_t)&ktile[waveu][0][0];
  koff[1] = (unsigned int)(uintptr_t)&ktile[waveu][1][0];
  voff[0] = (unsigned int)(uintptr_t)&vtile[waveu][0][0];
  voff[1] = (unsigned int)(uintptr_t)&vtile[waveu][1][0];

  // Preload Q A-fragments for all 16 k-steps of DH=64 (32 VGPRs).
  v2f qa[16];
  const float* Qrow = yq + ((size_t)(b * kS + m0 + l16)) * kE + h * kDH + hi2;
#pragma unroll
  for (int ks = 0; ks < 16; ++ks) {
    qa[ks].x = Qrow[ks * 4];
    qa[ks].y = Qrow[ks * 4 + 1];
  }

  v8f acc[4] = {};
  float mi[8], li[8];
#pragma unroll
  for (int v = 0; v < 8; ++v) {
    mi[v] = -__builtin_inff();
    li[v] = 0.0f;
  }

  const float* Kb = Kf + (size_t)b * kL * kDH;
  const float* Vb = Vf + (size_t)b * kL * kDH;
  float* pw = &plds[waveu][0];

  // Prime the pipeline: chunk 0 into buffer 0.
  tdm_load_2d(Kb, koff[0], 16, 64, 64, 5);
  tdm_load_2d(Vb, voff[0], 16, 64, 64, 5);

  int n = 0;
#pragma unroll 1
  for (int l0 = 0; l0 < kL; l0 += 16) {
    // Issue next chunk's DMAs into the other buffer (dummy reload at the end).
    const int ln = (l0 + 16 < kL) ? (l0 + 16) : 0;
    tdm_load_2d(Kb + (size_t)ln * kDH, koff[n ^ 1], 16, 64, 64, 5);
    tdm_load_2d(Vb + (size_t)ln * kDH, voff[n ^ 1], 16, 64, 64, 5);
    // Allow the 2 just-issued DMAs to stay in flight; older pair is done.
    __builtin_amdgcn_s_wait_tensorcnt(2);
    attn_chunk_lds(&ktile[waveu][n][0], &vtile[waveu][n][0], qa, pw, hi, l16,
                   acc, mi, li);
    n ^= 1;
  }

  // ---- finalize: divide by row sums, store as [B, S, H*DH]
#pragma unroll
  for (int v = 0; v < 8; ++v) {
    const float inv = 1.0f / li[v];
    const int r = m0 + v + hi * 8;
    float* Or = O + ((size_t)(b * kS + r)) * kE + h * kDH + l16;
    Or[0]  = acc[0][v] * inv;
    Or[16] = acc[1][v] * inv;
    Or[32] = acc[2][v] * inv;
    Or[48] = acc[3][v] * inv;
  }
}

// ---------------------------------------------------------------------------
extern "C" void kernel_launch(void* const* d_in, const int* in_sizes, int n_in,
                              void* d_out, int out_size, void* d_ws,
                              size_t ws_size, hipStream_t stream) {
  const float* x  = (const float*)d_in[0];
  const float* kc = (const float*)d_in[1];
  const float* vc = (const float*)d_in[2];
  const float* Wq = (const float*)d_in[3];
  const float* bq = (const float*)d_in[4];
  const float* Wk = (const float*)d_in[5];
  const float* bk = (const float*)d_in[6];
  const float* Wv = (const float*)d_in[7];
  const float* bv = (const float*)d_in[8];
  const float* Wo = (const float*)d_in[9];
  const float* bo = (const float*)d_in[10];
  float* out = (float*)d_out;

  // workspace layout (floats): 12 MB total
  float* ws = (float*)d_ws;
  const size_t nXE = (size_t)kB * kS * kE;   // 262144
  const size_t nKV = (size_t)kB * kL * kDH;  // 1048576
  float* yq   = ws;
  float* yk   = yq + nXE;
  float* yv   = yk + nXE;
  float* Kf   = yv + nXE;
  float* Vf   = Kf + nKV;
  float* attn = Vf + nKV;

  const int M = kB * kS;  // 512
  const int N = kE;       // 512
  const int K = kE;       // 512
  dim3 ggrid(M / 64, N / 64);
  dim3 gblock(128);

  // 1) QKV projections (TDM-fed WMMA GEMMs)
  gemm_xwt_wmma<<<ggrid, gblock, 0, stream>>>(x, Wq, bq, yq, M, N, K);
  gemm_xwt_wmma<<<ggrid, gblock, 0, stream>>>(x, Wk, bk, yk, M, N, K);
  gemm_xwt_wmma<<<ggrid, gblock, 0, stream>>>(x, Wv, bv, yv, M, N, K);

  // 2) K/V cache concat (contiguous [B, L, DH]), float4-vectorized
  {
    const int total4 = kB * kL * (kDH / 4);
    repack_kv<<<(total4 + 255) / 256, 256, 0, stream>>>(kc, vc, yk, yv, Kf, Vf);
  }

  // 3) TDM-fed flash attention: grid (S/16/2, H, B), 2 waves/block
  attn_flash_wmma<<<dim3(kS / 16 / 2, kH, kB), dim3(64), 0, stream>>>(
      yq, Kf, Vf, attn);

  // 4) output projection -> d_out
  gemm_xwt_wmma<<<ggrid, gblock, 0, stream>>>(attn, Wo, bo, out, M, N, K);
}